// PatchQuantumGenerator_21019569947163
// MI455X (gfx1250) — compile-verified
//
#include <hip/hip_runtime.h>
#include <math.h>

typedef __attribute__((ext_vector_type(2))) float v2f;
typedef __attribute__((ext_vector_type(8))) float v8f;

#define NQ     4
#define QD     6
#define NGEN   45
#define BOND   9
#define PATCH  5
#define UPPER  36
#define BATCH  32768

// ---------------------------------------------------------------------------
// Kernel 1: build the per-generator 16x16 circuit matrix, stored as the WMMA
// B operand (row-major B[k][s] = M_g[s][k]):  wsM[g*256 + k*16 + s]
// One block per generator; thread j simulates basis column |j>.
// ---------------------------------------------------------------------------
__global__ void build_unitaries(const float* __restrict__ W,
                                float* __restrict__ wsM) {
  const int g = blockIdx.x;
  const int j = threadIdx.x & 15;

  float st[16];
#pragma unroll
  for (int i = 0; i < 16; ++i) st[i] = (i == j) ? 1.0f : 0.0f;

  for (int layer = 0; layer < QD; ++layer) {
#pragma unroll
    for (int w = 0; w < NQ; ++w) {
      float th = 0.5f * W[(g * QD + layer) * NQ + w];
      float s, c;
      sincosf(th, &s, &c);
      const int mask = 8 >> w;   // wire w <-> bit (3-w)
#pragma unroll
      for (int i = 0; i < 16; ++i) {
        if ((i & mask) == 0) {
          const int i1 = i | mask;
          float a0 = st[i], a1 = st[i1];
          st[i]  = c * a0 - s * a1;
          st[i1] = s * a0 + c * a1;
        }
      }
    }
    // CZ between adjacent wires: sign flip where both bits set
#pragma unroll
    for (int w = 0; w < NQ - 1; ++w) {
      const int m2 = (8 >> w) | (8 >> (w + 1));
#pragma unroll
      for (int i = 0; i < 16; ++i)
        if ((i & m2) == m2) st[i] = -st[i];
    }
  }

  if (threadIdx.x < 16) {
#pragma unroll
    for (int s = 0; s < 16; ++s)
      wsM[g * 256 + j * 16 + s] = st[s];   // B row j: M_g[s][j]
  }
}

// ---------------------------------------------------------------------------
// Kernel 2: per-wave tile of 16 batch rows; for each generator g:
//   amps(16x16) = enc(16x16, k-major) x B_g  via 4x V_WMMA_F32_16X16X4_F32,
// then LDS transpose -> probs -> normalize -> softmax -> scatter.
// ---------------------------------------------------------------------------
__global__ __launch_bounds__(256)
void qgen_main(const float* __restrict__ X,
               const float* __restrict__ wsM,
               float* __restrict__ out) {
  __shared__ float lds[8][16][17];          // one slab per wave, padded

  const int lane = threadIdx.x & 31;
  const int wv   = threadIdx.x >> 5;
  const int half = lane >> 4;
  const int m    = lane & 15;               // batch row within tile / column
  const int b    = (blockIdx.x * 8 + wv) * 16 + m;

  // encoding vector enc[k] = prod_w (bit_w(k) ? sin(x_w/2) : cos(x_w/2))
  const float4 xv = ((const float4*)X)[b];
  float c0, s0, c1, s1, c2, s2, c3, s3;
  sincosf(0.5f * xv.x, &s0, &c0);
  sincosf(0.5f * xv.y, &s1, &c1);
  sincosf(0.5f * xv.z, &s2, &c2);
  sincosf(0.5f * xv.w, &s3, &c3);
  float p01[4] = {c0 * c1, c0 * s1, s0 * c1, s0 * s1};
  float p23[4] = {c2 * c3, c2 * s3, s2 * c3, s2 * s3};
  float enc[16];
#pragma unroll
  for (int k = 0; k < 16; ++k) enc[k] = p01[k >> 2] * p23[k & 3];

  // A fragments (16x4 f32): lanes 0-15 hold K=0/1, lanes 16-31 hold K=2/3
  v2f afrag[4];
#pragma unroll
  for (int kk = 0; kk < 4; ++kk) {
    afrag[kk].x = half ? enc[kk * 4 + 2] : enc[kk * 4 + 0];
    afrag[kk].y = half ? enc[kk * 4 + 3] : enc[kk * 4 + 1];
  }

  float* edges = out;
  float* nodes = out + (size_t)BATCH * 405;

  for (int g = 0; g < NGEN; ++g) {
    // B fragments (4x16 f32), analogous striping: VGPR0 rows K=0/2, VGPR1 K=1/3
    v2f bfrag[4];
#pragma unroll
    for (int kk = 0; kk < 4; ++kk) {
      const int k0 = kk * 4 + 2 * half;
      bfrag[kk].x = wsM[g * 256 + (k0    ) * 16 + m];
      bfrag[kk].y = wsM[g * 256 + (k0 + 1) * 16 + m];
    }

    v8f c = {};
#pragma unroll
    for (int kk = 0; kk < 4; ++kk)
      c = __builtin_amdgcn_wmma_f32_16x16x4_f32(
              false, afrag[kk], false, bfrag[kk],
              (short)0, c, false, false);

    // C layout: VGPR r, lanes0-15: (row r, col lane); lanes16-31: (row r+8)
#pragma unroll
    for (int r = 0; r < 8; ++r)
      lds[wv][r + 8 * half][m] = c[r];
    __syncthreads();

    float v[16];
#pragma unroll
    for (int s = 0; s < 16; ++s) v[s] = lds[wv][m][s];   // full row per lane
    __syncthreads();

    // probs -> p[0..4]/sum -> softmax
    float sum = 0.0f;
#pragma unroll
    for (int s = 0; s < 16; ++s) sum += v[s] * v[s];
    const float inv = 1.0f / sum;
    float p[PATCH];
#pragma unroll
    for (int jj = 0; jj < PATCH; ++jj) p[jj] = v[jj] * v[jj] * inv;
    float mx = p[0];
#pragma unroll
    for (int jj = 1; jj < PATCH; ++jj) mx = fmaxf(mx, p[jj]);
    float e[PATCH];
    float se = 0.0f;
#pragma unroll
    for (int jj = 0; jj < PATCH; ++jj) { e[jj] = __expf(p[jj] - mx); se += e[jj]; }
    const float rse = 1.0f / se;

    if (g < UPPER) {
      // triu index g -> (r, cc); half 0 writes (r,cc), half 1 writes (cc,r)
      int r = 0, rem = g, span = BOND - 1;
      while (rem >= span) { rem -= span; --span; ++r; }
      const int cc = r + 1 + rem;
      const size_t base = half
          ? ((size_t)b * 81 + (size_t)cc * 9 + r) * PATCH
          : ((size_t)b * 81 + (size_t)r * 9 + cc) * PATCH;
#pragma unroll
      for (int jj = 0; jj < PATCH; ++jj) edges[base + jj] = e[jj] * rse;
    } else if (half == 0) {
      const int n = g - UPPER;
      const size_t base = ((size_t)b * BOND + n) * PATCH;
#pragma unroll
      for (int jj = 0; jj < PATCH; ++jj) nodes[base + jj] = e[jj] * rse;
    }
  }
}

// ---------------------------------------------------------------------------
// Kernel 3: constant GT tensors + zero the edges diagonal.
// ---------------------------------------------------------------------------
__global__ void fill_const(float* __restrict__ out) {
  const size_t idx = (size_t)blockIdx.x * blockDim.x + threadIdx.x;
  const size_t NE = (size_t)BATCH * 405;
  const size_t NN = (size_t)BATCH * 45;
  float* edges    = out;
  float* edges_gt = out + NE + NN;
  float* nodes_gt = edges_gt + NE;

  if (idx < NE) {
    const int rem = (int)(idx % 405);
    const int rc = rem / 5, p = rem % 5;
    const int r = rc / 9, cdx = rc % 9;
    edges_gt[idx] = (r != cdx && p == 4) ? 1.0f : 0.0f;
  }
  if (idx < NN) {
    const int rem = (int)(idx % 45);
    const int p = rem % 5;
    nodes_gt[idx] = (p == 4) ? 1.0f : 0.0f;
    const size_t bb = idx / 45;
    const int r = rem / 5;
    edges[(bb * 81 + (size_t)r * 10) * 5 + p] = 0.0f;   // (r,r) diagonal
  }
}

// ---------------------------------------------------------------------------
extern "C" void kernel_launch(void* const* d_in, const int* in_sizes, int n_in,
                              void* d_out, int out_size, void* d_ws, size_t ws_size,
                              hipStream_t stream) {
  const float* x   = (const float*)d_in[0];   // (32768, 4) f32
  const float* wts = (const float*)d_in[1];   // (45, 6, 4) f32
  float* out = (float*)d_out;                 // edges|nodes|edges_GT|nodes_GT
  float* wsM = (float*)d_ws;                  // 45*256 f32 = 46 KB

  build_unitaries<<<NGEN, 32, 0, stream>>>(wts, wsM);
  qgen_main<<<BATCH / 128, 256, 0, stream>>>(x, wsM, out);

  const size_t nfill = (size_t)BATCH * 405;
  fill_const<<<(unsigned)((nfill + 255) / 256), 256, 0, stream>>>(out);
}